// RingDilatedAttentionCorrect_65755949301789
// MI455X (gfx1250) — compile-verified
//
#include <hip/hip_runtime.h>
#include <math.h>

typedef __attribute__((ext_vector_type(2))) float v2f;
typedef __attribute__((ext_vector_type(8))) float v8f;
typedef __attribute__((ext_vector_type(4))) unsigned int u32x4;
typedef __attribute__((ext_vector_type(8))) int i32x8;
typedef __attribute__((ext_vector_type(4))) int i32x4;

__device__ __forceinline__ v8f wmma4(v2f a, v2f b, v8f c) {
  // D = A(16x4,f32) * B(4x16,f32) + C(16x16,f32)
  return __builtin_amdgcn_wmma_f32_16x16x4_f32(false, a, false, b, (short)0, c,
                                               false, false);
}

// Async copy 16B global -> LDS (per-lane addresses), tracked by ASYNCcnt.
// VDST VGPR holds the LDS byte offset = low 32 bits of the flat shared ptr.
__device__ __forceinline__ void async_b128(const float* g, const void* lds) {
  unsigned l = (unsigned)(uintptr_t)lds;
  unsigned long long ga = (unsigned long long)(uintptr_t)g;
  asm volatile("global_load_async_to_lds_b128 %0, %1, off"
               :: "v"(l), "v"(ga) : "memory");
}
__device__ __forceinline__ void wait_async0() {
  asm volatile("s_wait_asynccnt 0" ::: "memory");
}

// TDM: one wave-level DMA of a 2D tile (64 rows x 64 f32, global row stride
// `stride` elems) into LDS with a 4-DWORD pad after every 64 DWORDs
// (=> LDS row stride 68 floats). Descriptor per CDNA5 ISA ch.8:
//  group0: count=1 | lds_addr | global_addr[56:0] | type=2
//  group1: data_size=2(4B), pad_enable, pad_interval=5(64 DW), pad_amount=3(4 DW),
//          tensor_dim0/1 = tile_dim0/1 = 64, tensor_dim0_stride = stride
__device__ __forceinline__ void tdm_load_k64x64(const void* gptr,
                                                unsigned lds_off,
                                                unsigned stride) {
  unsigned long long ga = (unsigned long long)(uintptr_t)gptr;
  u32x4 g0 = {1u, lds_off, (unsigned)ga,
              (unsigned)((ga >> 32) & 0x1FFFFFFu) | (2u << 30)};
  i32x8 g1 = {(int)((2u << 16) | (1u << 20) | (5u << 22) | (3u << 25)),
              (int)(64u << 16),   // tensor_dim0 = 64
              (int)(64u << 16),   // tensor_dim1 = 64
              (int)(64u << 16),   // tile_dim0   = 64
              (int)64,            // tile_dim1   = 64 (tile_dim2 = 0)
              (int)stride,        // tensor_dim0_stride[31:0]
              0, 0};
  i32x4 z4 = {0, 0, 0, 0};
  i32x8 z8 = {0, 0, 0, 0, 0, 0, 0, 0};
  // 6-arg form (clang-23 / therock-10.0 headers): (g0, g1, g2, g3, extra, cpol)
  __builtin_amdgcn_tensor_load_to_lds(g0, g1, z4, z4, z8, 0);
}

// ---------------------------------------------------------------------------
// C[M,N] = A[M,K] @ W[N,K]^T + bias[N]   (f32; K mult of 16, M mult of 128,
// N mult of 64). Block = 256 threads = 8 waves; block tile 128x64.
// Wave w computes rows w*16..w*16+15, all 64 cols (4 WMMA accumulators).
// LDS is double-buffered and filled with async-to-LDS B128 copies so the
// k-block i+1 transfer overlaps the k-block i WMMAs (1 barrier / iter).
// ---------------------------------------------------------------------------
__global__ __launch_bounds__(256)
void gemm_xwt_bias(const float* __restrict__ A, const float* __restrict__ W,
                   const float* __restrict__ bias, float* __restrict__ C,
                   int M, int N, int K) {
  constexpr int BK = 16;
  constexpr int LP = BK + 4;            // 20 floats = 80B row: keeps 16B align
  __shared__ float As[2][128][LP];
  __shared__ float Ws[2][64][LP];

  const int tid  = threadIdx.x;
  const int wave = tid >> 5;
  const int lane = tid & 31;
  const int hf   = lane >> 4;           // k-pair select
  const int ln   = lane & 15;           // row/col within 16
  const int m0 = blockIdx.y * 128;
  const int n0 = blockIdx.x * 64;
  const int wm = wave << 4;             // wave row offset 0..112
  const int ar = tid >> 1;              // 0..127 A row loaded by this thread
  const int ac = (tid & 1) << 3;        // 0 or 8
  const int wr = tid >> 2;              // 0..63  W row
  const int wc = (tid & 3) << 2;        // 0,4,8,12

  const float* ga = A + (size_t)(m0 + ar) * K + ac;
  const float* gw = W + (size_t)(n0 + wr) * K + wc;

  v8f acc0 = {}, acc1 = {}, acc2 = {}, acc3 = {};

  auto issue = [&](int buf, int k0) {
    async_b128(ga + k0,     &As[buf][ar][ac]);
    async_b128(ga + k0 + 4, &As[buf][ar][ac + 4]);
    async_b128(gw + k0,     &Ws[buf][wr][wc]);
  };
  auto compute = [&](int buf) {
#pragma unroll
    for (int kk = 0; kk < BK; kk += 4) {
      const int ko = kk + 2 * hf;       // lanes<16: k,k+1 ; lanes>=16: k+2,k+3
      v2f a  = *(const v2f*)&As[buf][wm + ln][ko];
      v2f b0 = *(const v2f*)&Ws[buf][ 0 + ln][ko];
      v2f b1 = *(const v2f*)&Ws[buf][16 + ln][ko];
      v2f b2 = *(const v2f*)&Ws[buf][32 + ln][ko];
      v2f b3 = *(const v2f*)&Ws[buf][48 + ln][ko];
      acc0 = wmma4(a, b0, acc0);
      acc1 = wmma4(a, b1, acc1);
      acc2 = wmma4(a, b2, acc2);
      acc3 = wmma4(a, b3, acc3);
    }
  };

  // prologue: fill buffer 0
  issue(0, 0);
  wait_async0();
  __syncthreads();

  int p = 0;
  for (int k0 = 0; k0 < K - BK; k0 += BK) {   // steady state: always prefetch
    issue(p ^ 1, k0 + BK);
    compute(p);
    wait_async0();                      // next buffer landed (this wave's copies)
    __syncthreads();                    // everyone's copies + reads done
    p ^= 1;
  }
  compute(p);                           // last k-block: no prefetch

  // D layout: VGPR i -> row = i + 8*hf, col = ln (within 16x16 tile)
  const int row0 = m0 + wm + hf * 8;
  const int c0   = n0 + ln;
  const float bv0 = bias[c0];
  const float bv1 = bias[c0 + 16];
  const float bv2 = bias[c0 + 32];
  const float bv3 = bias[c0 + 48];
#pragma unroll
  for (int i = 0; i < 8; ++i) {
    float* crow = C + (size_t)(row0 + i) * N + c0;
    crow[0]  = acc0[i] + bv0;
    crow[16] = acc1[i] + bv1;
    crow[32] = acc2[i] + bv2;
    crow[48] = acc3[i] + bv3;
  }
}

// ---------------------------------------------------------------------------
// Block-diagonal attention. qkv layout: [B=2, S=4096, 3*1024] where
// Q = +h*64, K = +1024+h*64, V = +2048+h*64. Each block handles 128 query
// rows of one (b, h, seg); each wave owns 16 query rows x full kv width.
// Flash-style online softmax over 16 kv tiles of 64. The K tile is staged
// by ONE TDM tensor_load (wave 0, TENSORcnt); V goes through registers
// (transposed on store into Vt).
// ---------------------------------------------------------------------------
__global__ __launch_bounds__(256)
void seg_attn(const float* __restrict__ qkv, float* __restrict__ out) {
  __shared__ float Ks[64][68];      // [kv][d]  (272B row; TDM pads 64+4)
  __shared__ float Vt[64][68];      // [d][kv]
  __shared__ float Ps[8][16][68];   // per-wave P round-trip D->A layout

  const int tid  = threadIdx.x;
  const int wave = tid >> 5;
  const int lane = tid & 31;
  const int hf   = lane >> 4;
  const int ln   = lane & 15;
  const int qc  = blockIdx.x;        // 0..7 query chunk
  const int seg = blockIdx.y;        // 0..3
  const int b   = blockIdx.z >> 4;   // 0..1
  const int h   = blockIdx.z & 15;   // 0..15

  const float* base = qkv + (size_t)(b * 4096 + seg * 1024) * 3072 + h * 64;
  const float* Kb = base + 1024;
  const float* Vb = base + 2048;
  const int q0 = qc * 128 + wave * 16;
  const unsigned ks_lds = (unsigned)(uintptr_t)&Ks[0][0];

  // Q fragments in registers (A-layout)
  v2f qa[16];
  {
    const float* qp = base + (size_t)(q0 + ln) * 3072 + 2 * hf;
#pragma unroll
    for (int t = 0; t < 16; ++t) qa[t] = *(const v2f*)(qp + 4 * t);
  }

  v8f o0 = {}, o1 = {}, o2 = {}, o3 = {};
  float mr[8], lsum[8];
#pragma unroll
  for (int i = 0; i < 8; ++i) { mr[i] = -INFINITY; lsum[i] = 0.0f; }

  const int lkv = tid >> 2;        // 0..63 kv row loaded by this thread
  const int lc  = (tid & 3) << 4;  // 0,16,32,48 d-offset

  for (int kt = 0; kt < 16; ++kt) {
    const int kv0 = kt * 64;
    const float* vp = Vb + (size_t)(kv0 + lkv) * 3072 + lc;
    float4 v4[4];
#pragma unroll
    for (int j = 0; j < 4; ++j) v4[j] = *(const float4*)(vp + 4 * j);
    __syncthreads();  // previous tile's LDS reads complete
    if (wave == 0)    // one wave-level DMA for the whole 64x64 K tile
      tdm_load_k64x64(Kb + (size_t)kv0 * 3072, ks_lds, 3072u);
#pragma unroll
    for (int j = 0; j < 4; ++j) {
      Vt[lc + 4 * j + 0][lkv] = v4[j].x;
      Vt[lc + 4 * j + 1][lkv] = v4[j].y;
      Vt[lc + 4 * j + 2][lkv] = v4[j].z;
      Vt[lc + 4 * j + 3][lkv] = v4[j].w;
    }
    if (wave == 0) __builtin_amdgcn_s_wait_tensorcnt(0);
    __syncthreads();

    // S = Q @ K^T over d=64 (16 k-steps), 4 column subtiles of 16 kv
    v8f s0 = {}, s1 = {}, s2 = {}, s3 = {};
#pragma unroll
    for (int t = 0; t < 16; ++t) {
      const int ko = 4 * t + 2 * hf;
      v2f a = qa[t];
      s0 = wmma4(a, *(const v2f*)&Ks[ 0 + ln][ko], s0);
      s1 = wmma4(a, *(const v2f*)&Ks[16 + ln][ko], s1);
      s2 = wmma4(a, *(const v2f*)&Ks[32 + ln][ko], s2);
      s3 = wmma4(a, *(const v2f*)&Ks[48 + ln][ko], s3);
    }

    // Online softmax. Slot i holds row i + 8*hf; rows live across the 16
    // lanes of each half-wave, so xor-shuffles with masks 1..8 reduce rows.
    float mn[8], al[8], rs[8];
#pragma unroll
    for (int i = 0; i < 8; ++i) {
      s0[i] *= 0.125f; s1[i] *= 0.125f; s2[i] *= 0.125f; s3[i] *= 0.125f;
      mn[i] = fmaxf(fmaxf(s0[i], s1[i]), fmaxf(s2[i], s3[i]));
    }
#pragma unroll
    for (int off = 1; off < 16; off <<= 1)
#pragma unroll
      for (int i = 0; i < 8; ++i)
        mn[i] = fmaxf(mn[i], __shfl_xor(mn[i], off, 32));
#pragma unroll
    for (int i = 0; i < 8; ++i) {
      mn[i] = fmaxf(mn[i], mr[i]);
      al[i] = __expf(mr[i] - mn[i]);
      mr[i] = mn[i];
      s0[i] = __expf(s0[i] - mn[i]);
      s1[i] = __expf(s1[i] - mn[i]);
      s2[i] = __expf(s2[i] - mn[i]);
      s3[i] = __expf(s3[i] - mn[i]);
      rs[i] = (s0[i] + s1[i]) + (s2[i] + s3[i]);
    }
#pragma unroll
    for (int off = 1; off < 16; off <<= 1)
#pragma unroll
      for (int i = 0; i < 8; ++i)
        rs[i] += __shfl_xor(rs[i], off, 32);
#pragma unroll
    for (int i = 0; i < 8; ++i) {
      lsum[i] = lsum[i] * al[i] + rs[i];
      o0[i] *= al[i]; o1[i] *= al[i]; o2[i] *= al[i]; o3[i] *= al[i];
    }

    // P: D-layout -> per-wave LDS slab -> A-layout (same-wave LDS is in-order)
#pragma unroll
    for (int i = 0; i < 8; ++i) {
      const int r = i + hf * 8;
      Ps[wave][r][ 0 + ln] = s0[i];
      Ps[wave][r][16 + ln] = s1[i];
      Ps[wave][r][32 + ln] = s2[i];
      Ps[wave][r][48 + ln] = s3[i];
    }
    // O += P @ V   (kv=64 contraction = 16 k-steps; 4 output d-subtiles)
#pragma unroll
    for (int t = 0; t < 16; ++t) {
      const int ko = 4 * t + 2 * hf;
      v2f a = *(const v2f*)&Ps[wave][ln][ko];
      o0 = wmma4(a, *(const v2f*)&Vt[ 0 + ln][ko], o0);
      o1 = wmma4(a, *(const v2f*)&Vt[16 + ln][ko], o1);
      o2 = wmma4(a, *(const v2f*)&Vt[32 + ln][ko], o2);
      o3 = wmma4(a, *(const v2f*)&Vt[48 + ln][ko], o3);
    }
  }

  // Normalize and store to [B, S, E] with col = h*64 + d
  float* ob = out + (size_t)(b * 4096 + seg * 1024 + q0) * 1024 + h * 64 + ln;
#pragma unroll
  for (int i = 0; i < 8; ++i) {
    const int r = i + hf * 8;
    const float inv = 1.0f / lsum[i];
    float* orow = ob + (size_t)r * 1024;
    orow[0]  = o0[i] * inv;
    orow[16] = o1[i] * inv;
    orow[32] = o2[i] * inv;
    orow[48] = o3[i] * inv;
  }
}

// ---------------------------------------------------------------------------
extern "C" void kernel_launch(void* const* d_in, const int* in_sizes, int n_in,
                              void* d_out, int out_size, void* d_ws,
                              size_t ws_size, hipStream_t stream) {
  (void)in_sizes; (void)n_in; (void)out_size; (void)ws_size;
  const float* x     = (const float*)d_in[0];
  const float* w_qkv = (const float*)d_in[1];
  const float* b_qkv = (const float*)d_in[2];
  const float* w_out = (const float*)d_in[3];
  const float* b_out = (const float*)d_in[4];
  float* out = (float*)d_out;

  float* qkv  = (float*)d_ws;               // [8192, 3072] f32 (100.7 MB)
  float* attn = qkv + (size_t)8192 * 3072;  // [8192, 1024] f32 (33.6 MB)

  dim3 blk(256);
  // 1) QKV projection: [8192,1024] @ [1024,3072]
  gemm_xwt_bias<<<dim3(3072 / 64, 8192 / 128), blk, 0, stream>>>(
      x, w_qkv, b_qkv, qkv, 8192, 3072, 1024);
  // 2) block-diagonal attention: grid (qchunk=8, seg=4, b*H=32)
  seg_attn<<<dim3(8, 4, 32), blk, 0, stream>>>(qkv, attn);
  // 3) output projection: [8192,1024] @ [1024,1024]
  gemm_xwt_bias<<<dim3(1024 / 64, 8192 / 128), blk, 0, stream>>>(
      attn, w_out, b_out, out, 8192, 1024, 1024);
}